// ResGCN_43456479101294
// MI455X (gfx1250) — compile-verified
//
#include <hip/hip_runtime.h>
#include <hip/hip_bf16.h>

typedef __attribute__((ext_vector_type(2))) float v2f;
typedef __attribute__((ext_vector_type(8))) float v8f;

#define NFEAT 64
#define EPS 1e-5f

// ---------------- degree / norm ----------------
__global__ void k_fill1(float* __restrict__ p, int n) {
    int i = blockIdx.x * blockDim.x + threadIdx.x;
    if (i < n) p[i] = 1.0f;  // self loop contributes 1 to every degree
}

__global__ void k_count_deg(const int* __restrict__ dst, float* __restrict__ deg, int nE) {
    int e = blockIdx.x * blockDim.x + threadIdx.x;
    if (e < nE) atomicAdd(&deg[dst[e]], 1.0f);
}

__global__ void k_rsqrt_inplace(float* __restrict__ p, int n) {
    int i = blockIdx.x * blockDim.x + threadIdx.x;
    if (i < n) p[i] = rsqrtf(p[i]);
}

// ---------------- WMMA GEMM: Y[n,64] = X[n,64] @ W[64,64] ----------------
// one wave per 16-row tile; 4 v8f accumulators cover all 64 output columns.
__global__ void k_gemm64_wmma(const float* __restrict__ X, const float* __restrict__ W,
                              float* __restrict__ Y, int nrows) {
    int wave = (blockIdx.x * blockDim.x + threadIdx.x) >> 5;  // global wave id
    int lane = threadIdx.x & 31;
    int row0 = wave * 16;
    if (row0 >= nrows) return;               // wave-uniform
    int m  = lane & 15;                      // A row / B,C column within tile
    int hi = lane >> 4;                      // K half select (A: K+2*hi, B: K+2*hi)
    const float* xrow = X + (size_t)(row0 + m) * NFEAT + 2 * hi;

    v8f acc[4] = {v8f{}, v8f{}, v8f{}, v8f{}};
#pragma unroll
    for (int k = 0; k < 64; k += 4) {
        v2f a = *(const v2f*)(xrow + k);     // X[row, k+2*hi], X[row, k+2*hi+1]
        int kk = k + 2 * hi;
#pragma unroll
        for (int t = 0; t < 4; ++t) {
            int col = t * 16 + m;
            v2f b;
            b[0] = W[kk * NFEAT + col];      // W[K, col]
            b[1] = W[(kk + 1) * NFEAT + col];
            acc[t] = __builtin_amdgcn_wmma_f32_16x16x4_f32(
                false, a, false, b, (short)0, acc[t], false, false);
        }
    }
#pragma unroll
    for (int t = 0; t < 4; ++t)
#pragma unroll
        for (int v = 0; v < 8; ++v)
            Y[(size_t)(row0 + v + 8 * hi) * NFEAT + t * 16 + m] = acc[t][v];
}

// ---------------- aggregation ----------------
// agg[i,:] = b[:] + h[i,:] * dinv[i]^2   (self-loop term; also initializes the buffer)
__global__ void k_agg_init(const float* __restrict__ h, const float* __restrict__ bias,
                           const float* __restrict__ dinv, float* __restrict__ agg, int n) {
    int i = blockIdx.x * blockDim.x + threadIdx.x;
    if (i >= n * NFEAT) return;
    int row = i >> 6, f = i & 63;
    float w = dinv[row];
    agg[i] = bias[f] + h[i] * w * w;
}

// one wave per edge; each lane scatters two features
__global__ void k_scatter64(const int* __restrict__ src, const int* __restrict__ dst,
                            const float* __restrict__ dinv, const float* __restrict__ h,
                            float* __restrict__ agg, int nE) {
    int e = (blockIdx.x * blockDim.x + threadIdx.x) >> 5;
    if (e >= nE) return;
    int lane = threadIdx.x & 31;
    int s = src[e], d = dst[e];
    float w = dinv[s] * dinv[d];
    v2f v = *(const v2f*)(h + (size_t)s * NFEAT + 2 * lane);
    float* ap = agg + (size_t)d * NFEAT + 2 * lane;
    atomicAdd(ap, v[0] * w);
    atomicAdd(ap + 1, v[1] * w);
}

// ---------------- batchnorm ----------------
__global__ void k_zero(float* __restrict__ p, int n) {
    int i = blockIdx.x * blockDim.x + threadIdx.x;
    if (i < n) p[i] = 0.0f;
}

// per-column sum / sumsq; stride is a multiple of 64 so each thread owns one column
__global__ void k_stats(const float* __restrict__ a, float* __restrict__ stats, int total) {
    __shared__ float s1[256], s2[256];
    float a1 = 0.f, a2 = 0.f;
    int stride = gridDim.x * 256;
    for (int i = blockIdx.x * 256 + threadIdx.x; i < total; i += stride) {
        float v = a[i];
        a1 += v;
        a2 += v * v;
    }
    s1[threadIdx.x] = a1;
    s2[threadIdx.x] = a2;
    __syncthreads();
    if (threadIdx.x < 64) {
        float t1 = s1[threadIdx.x] + s1[threadIdx.x + 64] + s1[threadIdx.x + 128] + s1[threadIdx.x + 192];
        float t2 = s2[threadIdx.x] + s2[threadIdx.x + 64] + s2[threadIdx.x + 128] + s2[threadIdx.x + 192];
        atomicAdd(&stats[threadIdx.x], t1);
        atomicAdd(&stats[64 + threadIdx.x], t2);
    }
}

// stats[f] -> mean, stats[64+f] -> rsqrt(var+eps)
__global__ void k_stats_finalize(float* __restrict__ stats, float invN) {
    int f = threadIdx.x;
    if (f >= 64) return;
    float mean = stats[f] * invN;
    float var = stats[64 + f] * invN - mean * mean;
    stats[f] = mean;
    stats[64 + f] = rsqrtf(var + EPS);
}

// out = relu((agg-mean)*rsig*g + be) [+ res]
__global__ void k_bn_relu(const float* __restrict__ agg, const float* __restrict__ stats,
                          const float* __restrict__ g, const float* __restrict__ be,
                          const float* __restrict__ res, float* __restrict__ out, int n) {
    int i = blockIdx.x * blockDim.x + threadIdx.x;
    if (i >= n * NFEAT) return;
    int f = i & 63;
    float v = (agg[i] - stats[f]) * stats[64 + f] * g[f] + be[f];
    v = fmaxf(v, 0.0f);
    if (res) v += res[i];
    out[i] = v;
}

// ---------------- final narrow layer: h2[n,2] = h[n,64] @ W2[64,2] ----------------
__global__ void k_gemm2(const float* __restrict__ h, const float* __restrict__ W2,
                        float* __restrict__ h2, int n) {
    int r = blockIdx.x * blockDim.x + threadIdx.x;
    if (r >= n) return;
    const float* hr = h + (size_t)r * NFEAT;
    float a0 = 0.f, a1 = 0.f;
#pragma unroll
    for (int f = 0; f < NFEAT; ++f) {
        float v = hr[f];
        a0 = fmaf(v, W2[2 * f], a0);
        a1 = fmaf(v, W2[2 * f + 1], a1);
    }
    h2[2 * r] = a0;
    h2[2 * r + 1] = a1;
}

__global__ void k_out_init(const float* __restrict__ h2, const float* __restrict__ b2,
                           const float* __restrict__ dinv, float* __restrict__ out, int n) {
    int i = blockIdx.x * blockDim.x + threadIdx.x;
    if (i >= n) return;
    float w = dinv[i];
    w = w * w;
    out[2 * i]     = b2[0] + h2[2 * i] * w;
    out[2 * i + 1] = b2[1] + h2[2 * i + 1] * w;
}

__global__ void k_scatter2(const int* __restrict__ src, const int* __restrict__ dst,
                           const float* __restrict__ dinv, const float* __restrict__ h2,
                           float* __restrict__ out, int nE) {
    int e = blockIdx.x * blockDim.x + threadIdx.x;
    if (e >= nE) return;
    int s = src[e], d = dst[e];
    float w = dinv[s] * dinv[d];
    atomicAdd(&out[2 * d], h2[2 * s] * w);
    atomicAdd(&out[2 * d + 1], h2[2 * s + 1] * w);
}

extern "C" void kernel_launch(void* const* d_in, const int* in_sizes, int n_in,
                              void* d_out, int out_size, void* d_ws, size_t ws_size,
                              hipStream_t stream) {
    const float* x   = (const float*)d_in[0];
    const int*   ei  = (const int*)d_in[1];
    const float* W0  = (const float*)d_in[2];
    const float* b0  = (const float*)d_in[3];
    const float* g0  = (const float*)d_in[4];
    const float* be0 = (const float*)d_in[5];
    const float* W1  = (const float*)d_in[6];
    const float* b1  = (const float*)d_in[7];
    const float* g1  = (const float*)d_in[8];
    const float* be1 = (const float*)d_in[9];
    const float* W2  = (const float*)d_in[10];
    const float* b2  = (const float*)d_in[11];
    float* out = (float*)d_out;

    const int nN = in_sizes[0] / NFEAT;   // 100000
    const int nE = in_sizes[1] / 2;       // 1600000
    const int* src = ei;
    const int* dst = ei + nE;

    // workspace layout (floats)
    float* ws    = (float*)d_ws;
    float* dinv  = ws;                          // nN
    float* h     = dinv + nN;                   // nN*64
    float* agg   = h + (size_t)nN * NFEAT;      // nN*64
    float* hres  = agg + (size_t)nN * NFEAT;    // nN*64
    float* stats = hres + (size_t)nN * NFEAT;   // 128
    float* h2    = stats + 128;                 // nN*2

    const int TB = 256;
    int blkN   = (nN + TB - 1) / TB;
    int blkE   = (nE + TB - 1) / TB;
    int blkNF  = (nN * NFEAT + TB - 1) / TB;
    int waves  = nN / 16;                                 // 16-row tiles
    int blkGEM = (waves * 32 + TB - 1) / TB;
    int blkSC  = (nE * 32 + TB - 1) / TB;                 // wave per edge
    float invN = 1.0f / (float)nN;

    // degree -> dinv
    k_fill1<<<blkN, TB, 0, stream>>>(dinv, nN);
    k_count_deg<<<blkE, TB, 0, stream>>>(dst, dinv, nE);
    k_rsqrt_inplace<<<blkN, TB, 0, stream>>>(dinv, nN);

    // ---- layer 0 ----
    k_gemm64_wmma<<<blkGEM, TB, 0, stream>>>(x, W0, h, nN);
    k_agg_init<<<blkNF, TB, 0, stream>>>(h, b0, dinv, agg, nN);
    k_scatter64<<<blkSC, TB, 0, stream>>>(src, dst, dinv, h, agg, nE);
    k_zero<<<1, 128, 0, stream>>>(stats, 128);
    k_stats<<<512, TB, 0, stream>>>(agg, stats, nN * NFEAT);
    k_stats_finalize<<<1, 64, 0, stream>>>(stats, invN);
    k_bn_relu<<<blkNF, TB, 0, stream>>>(agg, stats, g0, be0, nullptr, hres, nN);

    // ---- layer 1 (+ residual) ----
    k_gemm64_wmma<<<blkGEM, TB, 0, stream>>>(hres, W1, h, nN);
    k_agg_init<<<blkNF, TB, 0, stream>>>(h, b1, dinv, agg, nN);
    k_scatter64<<<blkSC, TB, 0, stream>>>(src, dst, dinv, h, agg, nE);
    k_zero<<<1, 128, 0, stream>>>(stats, 128);
    k_stats<<<512, TB, 0, stream>>>(agg, stats, nN * NFEAT);
    k_stats_finalize<<<1, 64, 0, stream>>>(stats, invN);
    k_bn_relu<<<blkNF, TB, 0, stream>>>(agg, stats, g1, be1, hres, h, nN);

    // ---- layer 2 (64 -> 2) ----
    k_gemm2<<<blkN, TB, 0, stream>>>(h, W2, h2, nN);
    k_out_init<<<blkN, TB, 0, stream>>>(h2, b2, dinv, out, nN);
    k_scatter2<<<blkE, TB, 0, stream>>>(src, dst, dinv, h2, out, nE);
}